// TrajectoryGenerator_20701742366853
// MI455X (gfx1250) — compile-verified
//
#include <hip/hip_runtime.h>
#include <hip/hip_bf16.h>

// ---------------------------------------------------------------------------
// TrajectoryGenerator pool-net, MI455X (gfx1250, wave32, WMMA).
//
// Exact algebraic simplifications (bit-faithful up to fp reassociation):
//  * softmax over a singleton axis == 1.0  -> entire attention MLP is dead.
//  * velocity embedding is dead (reference discards it).
//  * weighted[b,i,j,:] = 0.05*feat_j[b,j,:] is i-independent -> the [B,P,P,*]
//    tensors collapse to [batch,*]; the final max over j is a per-group
//    column max broadcast to all P rows of the group.
// Surviving work: feat build, two fp32 GEMMs (1024x128x512, 1024x512x1024)
// with folded BN+ReLU epilogues, group-max pooling. ~1.2 GFLOP / ~13 MB moved
// -> memory-latency bound; fp32 WMMA (v_wmma_f32_16x16x4_f32) keeps full
// fp32 precision on the matrix pipes.
// ---------------------------------------------------------------------------

typedef __attribute__((ext_vector_type(2))) float v2f;
typedef __attribute__((ext_vector_type(8))) float v8f;

#define BATCH 1024
#define GROUPS 16
#define PED 64
#define FEAT_DIM 128   // H + E
#define HID1 512
#define HID2 1024

// feat[r, 0:64]  = 0.05 * h_states[r, :]
// feat[r, 64:128]= 0.05 * (agent_type[0,r] * colsum(wa_w) + wa_b)
__global__ void build_feat_kernel(const float* __restrict__ h,
                                  const float* __restrict__ at,
                                  const float* __restrict__ wa_w,
                                  const float* __restrict__ wa_b,
                                  float* __restrict__ feat) {
    int idx = blockIdx.x * blockDim.x + threadIdx.x;   // [0, 1024*128)
    int c = idx & (FEAT_DIM - 1);
    int r = idx >> 7;
    float val;
    if (c < 64) {
        val = h[r * 64 + c];
    } else {
        int e = c - 64;
        float ws = 0.0f;
#pragma unroll
        for (int k = 0; k < 6; ++k) ws += wa_w[k * 64 + e];
        val = at[r] * ws + wa_b[e];
    }
    feat[idx] = 0.05f * val;
}

// One 16x16 output tile per wave, accumulated with V_WMMA_F32_16X16X4_F32.
// A [M,K] row-major, W [K,N] row-major. Grid sized so every wave has a tile
// (EXEC all-1s at every WMMA, per ISA restriction).
// Epilogue: folded eval-mode BatchNorm + ReLU.
//   s  = gamma * rsqrt(var + eps)
//   sh = (bias - mean) * s + beta
__global__ void wmma_gemm_bn_relu_kernel(const float* __restrict__ A,
                                         const float* __restrict__ W,
                                         const float* __restrict__ bias,
                                         const float* __restrict__ bn, // [4,N]: g,b,m,v
                                         float* __restrict__ out,
                                         int M, int N, int K) {
    const int lane   = threadIdx.x & 31;
    const int wave   = blockIdx.x * (blockDim.x >> 5) + (threadIdx.x >> 5);
    const int tilesN = N >> 4;
    const int tm     = wave / tilesN;
    const int tn     = wave - tm * tilesN;

    // fp32 A-fragment (16x4): lane m = lane&15; VGPR v holds K = kb + v,
    // kb = 2*(lane>>4).  B-fragment (4x16): lane n = lane&15, same kb split.
    const int m  = lane & 15;
    const int n  = lane & 15;
    const int kb = (lane >> 4) << 1;

    const float* ap = A + (size_t)(tm * 16 + m) * K + kb;
    const float* bp = W + (size_t)kb * N + tn * 16 + n;

    v8f acc = {};
    for (int k0 = 0; k0 < K; k0 += 4) {
        v2f a = *(const v2f*)ap;          // A[m][kb], A[m][kb+1] (8B aligned)
        v2f b;
        b.x = bp[0];                      // W[kb][n]
        b.y = bp[N];                      // W[kb+1][n]
        ap += 4;
        bp += (size_t)4 * N;
        acc = __builtin_amdgcn_wmma_f32_16x16x4_f32(
            /*neg_a=*/false, a, /*neg_b=*/false, b,
            /*c_mod=*/(short)0, acc, /*reuse_a=*/false, /*reuse_b=*/false);
    }

    // C/D layout: VGPR v, lane L -> row = v + 8*(L>>4), col = L&15.
    const int col = tn * 16 + n;
    const float g  = bn[col];
    const float be = bn[N + col];
    const float mn = bn[2 * N + col];
    const float vr = bn[3 * N + col];
    const float s  = g * rsqrtf(vr + 1e-5f);
    const float sh = (bias[col] - mn) * s + be;

    const int rbase = tm * 16 + ((lane >> 4) << 3);
    float* op = out + (size_t)rbase * N + col;
#pragma unroll
    for (int v = 0; v < 8; ++v) {
        op[(size_t)v * N] = fmaxf(acc[v] * s + sh, 0.0f);
    }
}

// Per group b (64 rows), per column c: max over the 64 rows, broadcast back
// to all 64 output rows of the group. One thread per (b, c).
__global__ void pool_bcast_kernel(const float* __restrict__ act2,
                                  float* __restrict__ out) {
    int idx = blockIdx.x * blockDim.x + threadIdx.x;   // [0, 16*1024)
    int c = idx & (HID2 - 1);
    int b = idx >> 10;
    const float* base = act2 + (size_t)(b * PED) * HID2 + c;
    float mx = base[0];
#pragma unroll 8
    for (int j = 1; j < PED; ++j) mx = fmaxf(mx, base[(size_t)j * HID2]);
    float* ob = out + (size_t)(b * PED) * HID2 + c;
#pragma unroll 8
    for (int i = 0; i < PED; ++i) ob[(size_t)i * HID2] = mx;
}

extern "C" void kernel_launch(void* const* d_in, const int* in_sizes, int n_in,
                              void* d_out, int out_size, void* d_ws, size_t ws_size,
                              hipStream_t stream) {
    // setup_inputs() order
    const float* h_states   = (const float*)d_in[0];   // (1, 1024, 64)
    const float* agent_type = (const float*)d_in[4];   // (8, 1024, 1); [0] slice = first 1024
    const float* wa_w       = (const float*)d_in[9];   // (6, 64)
    const float* wa_b       = (const float*)d_in[10];  // (64,)
    const float* pre1_w     = (const float*)d_in[17];  // (128, 512)
    const float* pre1_b     = (const float*)d_in[18];  // (512,)
    const float* pre_bn1    = (const float*)d_in[19];  // (4, 512)
    const float* pre2_w     = (const float*)d_in[20];  // (512, 1024)
    const float* pre2_b     = (const float*)d_in[21];  // (1024,)
    const float* pre_bn2    = (const float*)d_in[22];  // (4, 1024)
    float* out = (float*)d_out;                        // (1024, 1024) fp32

    // Workspace: feat 512KB | act1 2MB | act2 4MB  (6.5MB total)
    char* ws = (char*)d_ws;
    float* feat = (float*)ws;
    float* act1 = (float*)(ws + (size_t)BATCH * FEAT_DIM * sizeof(float));
    float* act2 = (float*)(ws + (size_t)BATCH * (FEAT_DIM + HID1) * sizeof(float));

    // 1) feat: 1024*128 elements, 256 threads/block
    build_feat_kernel<<<(BATCH * FEAT_DIM) / 256, 256, 0, stream>>>(
        h_states, agent_type, wa_w, wa_b, feat);

    // 2) GEMM1: [1024,128] @ [128,512] -> act1; tiles = 64*32 = 2048; 8 waves/block
    wmma_gemm_bn_relu_kernel<<<(BATCH / 16) * (HID1 / 16) / 8, 256, 0, stream>>>(
        feat, pre1_w, pre1_b, pre_bn1, act1, BATCH, HID1, FEAT_DIM);

    // 3) GEMM2: [1024,512] @ [512,1024] -> act2; tiles = 64*64 = 4096
    wmma_gemm_bn_relu_kernel<<<(BATCH / 16) * (HID2 / 16) / 8, 256, 0, stream>>>(
        act1, pre2_w, pre2_b, pre_bn2, act2, BATCH, HID2, HID1);

    // 4) group max-pool + broadcast: 16*1024 threads
    pool_bcast_kernel<<<(GROUPS * HID2) / 256, 256, 0, stream>>>(act2, out);
}